// GHMR_Loss_7164005449995
// MI455X (gfx1250) — compile-verified
//
#include <hip/hip_runtime.h>

typedef __attribute__((ext_vector_type(2))) float v2f;
typedef __attribute__((ext_vector_type(8))) float v8f;

#define BINS 10
#define G_ALPHA 0.75f
#define G_MOMENTUM 0.9f
#define G_MU 1.0f

#define THREADS 256
#define HIST_BLOCKS 1024
#define LOSS_BLOCKS 1024

// ---- helpers ---------------------------------------------------------------

__device__ __forceinline__ int bin_index_from_ad(float ad) {
    // ad = |pred - target|, wrap to [0,180], normalize to [0,1], bin by 10
    float diff = fminf(ad, 360.0f - ad);
    float g = diff * (1.0f / 180.0f);
    int idx = (int)floorf(g * (float)BINS);
    idx = idx < 0 ? 0 : idx;
    idx = idx > (BINS - 1) ? (BINS - 1) : idx;
    return idx;
}

// ---- pass 1: histogram -----------------------------------------------------

__global__ void ghmr_hist_kernel(const float* __restrict__ pred,
                                 const float* __restrict__ target,
                                 unsigned int* __restrict__ gcounts, int n) {
    __shared__ unsigned int lhist[8 * BINS];  // per-wave privatized histograms
    const int wave = threadIdx.x >> 5;
    if (threadIdx.x < 8 * BINS) lhist[threadIdx.x] = 0u;
    __syncthreads();

    const int gid = blockIdx.x * blockDim.x + threadIdx.x;
    const int stride = gridDim.x * blockDim.x;
    const int n4 = n >> 2;
    const float4* p4 = (const float4*)pred;
    const float4* t4 = (const float4*)target;

    for (int i = gid; i < n4; i += stride) {
        float4 p = p4[i];
        float4 t = t4[i];
        atomicAdd(&lhist[wave * BINS + bin_index_from_ad(fabsf(p.x - t.x))], 1u);
        atomicAdd(&lhist[wave * BINS + bin_index_from_ad(fabsf(p.y - t.y))], 1u);
        atomicAdd(&lhist[wave * BINS + bin_index_from_ad(fabsf(p.z - t.z))], 1u);
        atomicAdd(&lhist[wave * BINS + bin_index_from_ad(fabsf(p.w - t.w))], 1u);
    }
    // scalar tail
    for (int i = (n4 << 2) + gid; i < n; i += stride) {
        atomicAdd(&lhist[wave * BINS + bin_index_from_ad(fabsf(pred[i] - target[i]))], 1u);
    }
    __syncthreads();

    if (threadIdx.x < BINS) {
        unsigned int s = 0;
        #pragma unroll
        for (int w = 0; w < 8; ++w) s += lhist[w * BINS + threadIdx.x];
        atomicAdd(&gcounts[threadIdx.x], s);  // integer atomics: deterministic
    }
}

// ---- pass 2: per-bin weights (tiny, one wave) ------------------------------

__global__ void ghmr_weights_kernel(const unsigned int* __restrict__ counts,
                                    const float* __restrict__ acc_sum,
                                    float* __restrict__ wfin, float total) {
    const int b = threadIdx.x;  // one wave32
    const float c = (b < BINS) ? (float)counts[b] : 0.0f;
    const float as = (b < BINS) ? acc_sum[b] : 0.0f;
    const float acc_new = (c > 0.0f) ? (G_MOMENTUM * as + (1.0f - G_MOMENTUM) * c) : as;
    const unsigned long long mask = __ballot((b < BINS) && (c > 0.0f));
    const float nf = (float)__popcll(mask);
    const float n_safe = fmaxf(nf, 1.0f);
    const float w = (acc_new > 0.0f) ? (total / (n_safe * fmaxf(acc_new, 1e-30f))) : 0.0f;
    const float wf = (nf > 0.0f) ? __powf(w, G_ALPHA) : 1.0f;
    if (b < BINS) wfin[b] = wf;
}

// ---- pass 3: weighted smooth-L1 accumulation + WMMA block reduction --------

__global__ void ghmr_loss_kernel(const float* __restrict__ pred,
                                 const float* __restrict__ target,
                                 const float* __restrict__ wfin,
                                 float* __restrict__ partials, int n) {
    __shared__ float sw[BINS];
    __shared__ float lanes[THREADS];
    if (threadIdx.x < BINS) sw[threadIdx.x] = wfin[threadIdx.x];
    __syncthreads();

    const int gid = blockIdx.x * blockDim.x + threadIdx.x;
    const int stride = gridDim.x * blockDim.x;
    const int n4 = n >> 2;
    const float4* p4 = (const float4*)pred;
    const float4* t4 = (const float4*)target;

    float acc = 0.0f;
    for (int i = gid; i < n4; i += stride) {
        float4 p = p4[i];
        float4 t = t4[i];
        {
            float d = p.x - t.x, ad = fabsf(d);
            float l = (ad < G_MU) ? (0.5f * d * d / G_MU) : (ad - 0.5f * G_MU);
            acc += l * sw[bin_index_from_ad(ad)];
        }
        {
            float d = p.y - t.y, ad = fabsf(d);
            float l = (ad < G_MU) ? (0.5f * d * d / G_MU) : (ad - 0.5f * G_MU);
            acc += l * sw[bin_index_from_ad(ad)];
        }
        {
            float d = p.z - t.z, ad = fabsf(d);
            float l = (ad < G_MU) ? (0.5f * d * d / G_MU) : (ad - 0.5f * G_MU);
            acc += l * sw[bin_index_from_ad(ad)];
        }
        {
            float d = p.w - t.w, ad = fabsf(d);
            float l = (ad < G_MU) ? (0.5f * d * d / G_MU) : (ad - 0.5f * G_MU);
            acc += l * sw[bin_index_from_ad(ad)];
        }
    }
    for (int i = (n4 << 2) + gid; i < n; i += stride) {
        float d = pred[i] - target[i], ad = fabsf(d);
        float l = (ad < G_MU) ? (0.5f * d * d / G_MU) : (ad - 0.5f * G_MU);
        acc += l * sw[bin_index_from_ad(ad)];
    }

    lanes[threadIdx.x] = acc;
    __syncthreads();

    // Block reduction on the matrix pipe: 256 partials -> 4x V_WMMA_F32_16X16X4_F32
    // A = 16x4 fp32 tile of partials (A layout: lanes 0-15 supply K=0,1 of row M=l;
    // lanes 16-31 supply K=2,3 of row M=l-16). B = all ones => D[m][n] = row-sum of A
    // accumulated over the 4 tiles, identical for every column n. Any lane pair
    // (l, l+16) then holds all 16 row sums across its 8 C VGPRs.
    if (threadIdx.x < 32) {  // wave 0 only: EXEC all-1s inside this wave
        const int l = threadIdx.x;
        const int m = (l < 16) ? l : (l - 16);
        const int k = (l < 16) ? 0 : 2;
        v2f ones; ones.x = 1.0f; ones.y = 1.0f;
        v8f c = {};
        #pragma unroll
        for (int j = 0; j < 4; ++j) {
            v2f a;
            a.x = lanes[j * 64 + m * 4 + k];
            a.y = lanes[j * 64 + m * 4 + k + 1];
            c = __builtin_amdgcn_wmma_f32_16x16x4_f32(
                false, a, false, ones, (short)0, c, false, false);
        }
        float s = c[0] + c[1] + c[2] + c[3] + c[4] + c[5] + c[6] + c[7];
        s += __shfl_xor(s, 16);  // combine rows 0-7 (lane l) with rows 8-15 (lane l+16)
        if (l == 0) partials[blockIdx.x] = s;
    }
}

// ---- pass 4: deterministic final reduction ---------------------------------

__global__ void ghmr_final_kernel(const float* __restrict__ partials, int nparts,
                                  float* __restrict__ out, float inv_total) {
    __shared__ float red[THREADS];
    float s = 0.0f;
    for (int i = threadIdx.x; i < nparts; i += THREADS) s += partials[i];
    red[threadIdx.x] = s;
    __syncthreads();
    #pragma unroll
    for (int off = THREADS / 2; off > 0; off >>= 1) {
        if (threadIdx.x < off) red[threadIdx.x] += red[threadIdx.x + off];
        __syncthreads();
    }
    if (threadIdx.x == 0) out[0] = red[0] * inv_total;
}

// ---- launch ----------------------------------------------------------------

extern "C" void kernel_launch(void* const* d_in, const int* in_sizes, int n_in,
                              void* d_out, int out_size, void* d_ws, size_t ws_size,
                              hipStream_t stream) {
    const float* pred    = (const float*)d_in[0];
    const float* target  = (const float*)d_in[1];
    const float* acc_sum = (const float*)d_in[2];
    float* out = (float*)d_out;
    const int n = in_sizes[0];

    // workspace layout
    unsigned int* counts = (unsigned int*)d_ws;                    // 10 u32 @ 0
    float* wfin          = (float*)((char*)d_ws + 64);             // 10 f32 @ 64
    float* partials      = (float*)((char*)d_ws + 128);            // LOSS_BLOCKS f32 @ 128

    hipMemsetAsync(d_ws, 0, 64, stream);  // zero counts every call (graph-safe)

    ghmr_hist_kernel<<<HIST_BLOCKS, THREADS, 0, stream>>>(pred, target, counts, n);
    ghmr_weights_kernel<<<1, 32, 0, stream>>>(counts, acc_sum, wfin, (float)n);
    ghmr_loss_kernel<<<LOSS_BLOCKS, THREADS, 0, stream>>>(pred, target, wfin, partials, n);
    ghmr_final_kernel<<<1, THREADS, 0, stream>>>(partials, LOSS_BLOCKS, out, 1.0f / (float)n);
}